// GaussianSelfAttention_46205258170941
// MI455X (gfx1250) — compile-verified
//
#include <hip/hip_runtime.h>

// GaussianSelfAttention, MI455X (gfx1250, wave32, WMMA).
//
// Key observation: softmax over a singleton axis == 1, so out = sampled_v.
// q/k GEMMs and the score/softmax are dead.  By linearity:
//   sample(x@Wv + bv) = (sample-combined x) @ Wv + (sum of weights) * bv
// so we pre-combine the 4 bilinear x-rows per patch and run ONE f32 GEMM
// (M=64*196=12544, N=768, K=768) via V_WMMA_F32_16X16X4_F32 (exact f32).

#define BB    64
#define SS    197
#define DD    768
#define PP    196
#define MM    (BB * PP)      // 12544 GEMM rows
#define GRIDF 14.0f

typedef float v2f __attribute__((ext_vector_type(2)));
typedef float v8f __attribute__((ext_vector_type(8)));

// ---------------------------------------------------------------------------
// Kernel 1: per (b,p) compute bilinear weights/indices (faithful to the
// reference: ceil/floor, trunc-toward-zero int cast, non-negative % 197),
// then xs[m,:] = w11*x[i11,:] + w21*x[i21,:] + w12*x[i12,:] + w22*x[i22,:]
// and sumw[m] = w11+w21+w12+w22.
// ---------------------------------------------------------------------------
__global__ void gsa_sample_combine(const float* __restrict__ x,
                                   const int*   __restrict__ img_ids,
                                   const float* __restrict__ avgs,
                                   const float* __restrict__ stds,
                                   const float* __restrict__ noise,
                                   float* __restrict__ xs,
                                   float* __restrict__ sumw)
{
    const int m = blockIdx.x;            // 0 .. MM-1
    const int b = m / PP;
    const int p = m - b * PP;
    const int img = img_ids[b];

    const float ax = avgs[(img * 2 + 0) * PP + p];
    const float ay = avgs[(img * 2 + 1) * PP + p];
    const float sx = stds[(img * 2 + 0) * PP + p];
    const float sy = stds[(img * 2 + 1) * PP + p];
    const float nx = noise[(b * 2 + 0) * PP + p];
    const float ny = noise[(b * 2 + 1) * PP + p];

    const float kx = (nx - ax) / sx;
    const float ky = (ny - ay) / sy;

    const float x1 = ceilf(kx),  x2 = floorf(kx);
    const float y1 = ceilf(ky),  y2 = floorf(ky);

    const float wx1 = 1.0f - fabsf(x1 - kx), wx2 = 1.0f - fabsf(x2 - kx);
    const float wy1 = 1.0f - fabsf(y1 - ky), wy2 = 1.0f - fabsf(y2 - ky);
    const float w11 = wx1 * wy1, w21 = wx2 * wy1;
    const float w12 = wx1 * wy2, w22 = wx2 * wy2;

    auto widx = [](float px, float py) -> int {
        int t = (int)(GRIDF * py + px);  // f32 math, trunc toward zero
        int r = t % SS;                  // python %: result sign of divisor
        return (r < 0) ? r + SS : r;
    };
    const int i11 = widx(x1, y1), i21 = widx(x2, y1);
    const int i12 = widx(x1, y2), i22 = widx(x2, y2);

    const float* __restrict__ xb  = x + (long)b * SS * DD;
    const float* __restrict__ r11 = xb + (long)i11 * DD;
    const float* __restrict__ r21 = xb + (long)i21 * DD;
    const float* __restrict__ r12 = xb + (long)i12 * DD;
    const float* __restrict__ r22 = xb + (long)i22 * DD;
    float* __restrict__ o = xs + (long)m * DD;

    for (int d = threadIdx.x; d < DD; d += blockDim.x)
        o[d] = w11 * r11[d] + w21 * r21[d] + w12 * r12[d] + w22 * r22[d];

    if (threadIdx.x == 0)
        sumw[m] = w11 + w21 + w12 + w22;
}

// ---------------------------------------------------------------------------
// Kernel 2: f32 WMMA GEMM.  out_row(m) = xs[m,:] @ Wv + sumw[m]*bv,
// scattered to out[b, 1+p, :].
// WG tile 64x64, 8 waves arranged 4(M) x 2(N); each wave owns two 16x16
// f32 accumulators (cols +0 and +16 inside its 32-col slab), K step = 32.
// Fragment layouts per CDNA5 ISA 7.12.2 (V_WMMA_F32_16X16X4_F32):
//   A 16x4 : lanes 0-15 = rows, {v0,v1} = K 0,1 ; lanes 16-31 = K 2,3
//   B 4x16 : lanes 0-15 = cols, {v0,v1} = K 0,1 ; lanes 16-31 = K 2,3
//   C/D    : vgpr r, lanes 0-15 -> (M=r, N=lane); lanes 16-31 -> (M=r+8).
// ---------------------------------------------------------------------------
#define BM 64
#define BN 64
#define BK 32

__global__ __launch_bounds__(256) void
gsa_gemm_wmma(const float* __restrict__ A,     // xs: MM x DD
              const float* __restrict__ Wv,    // DD x DD
              const float* __restrict__ bv,    // DD
              const float* __restrict__ sumw,  // MM
              float* __restrict__ out)         // BB x SS x DD
{
    __shared__ float As[BM][BK + 2];
    __shared__ float Bs[BK][BN + 2];

    const int tid  = threadIdx.x;
    const int lane = tid & 31;
    const int wave = tid >> 5;
    const int wm   = wave & 3;     // M sub-tile 0..3
    const int wn   = wave >> 2;    // N slab 0..1

    const int m0 = blockIdx.x * BM;
    const int n0 = blockIdx.y * BN;

    const int ml   = lane & 15;
    const int half = lane >> 4;

    v8f c0 = {};
    v8f c1 = {};

    for (int k0 = 0; k0 < DD; k0 += BK) {
        // Stage A (64x32) and B (32x64), coalesced 128B rows.
        #pragma unroll
        for (int i = tid; i < BM * BK; i += 256) {
            const int r = i >> 5, cc = i & 31;
            As[r][cc] = A[(long)(m0 + r) * DD + k0 + cc];
        }
        #pragma unroll
        for (int i = tid; i < BK * BN; i += 256) {
            const int r = i >> 6, cc = i & 63;
            Bs[r][cc] = Wv[(long)(k0 + r) * DD + n0 + cc];
        }
        __syncthreads();

        const int arow  = wm * 16 + ml;
        const int bcol0 = wn * 32 + ml;

        #pragma unroll
        for (int k = 0; k < BK; k += 4) {
            const int ka = k + 2 * half;
            v2f a;  a.x  = As[arow][ka];       a.y  = As[arow][ka + 1];
            v2f b0; b0.x = Bs[ka][bcol0];      b0.y = Bs[ka + 1][bcol0];
            v2f b1; b1.x = Bs[ka][bcol0 + 16]; b1.y = Bs[ka + 1][bcol0 + 16];
            c0 = __builtin_amdgcn_wmma_f32_16x16x4_f32(
                     false, a, false, b0, (short)0, c0, false, false);
            c1 = __builtin_amdgcn_wmma_f32_16x16x4_f32(
                     false, a, false, b1, (short)0, c1, false, false);
        }
        __syncthreads();
    }

    // Epilogue: add sumw*bv, scatter row m -> out[b, 1+p, :].
    const int mbase = m0 + wm * 16 + half * 8;
    const int col0  = n0 + wn * 32 + ml;
    const int col1  = col0 + 16;
    const float bv0 = bv[col0];
    const float bv1 = bv[col1];

    #pragma unroll
    for (int r = 0; r < 8; ++r) {
        const int m  = mbase + r;
        const int b  = m / PP;
        const int p  = m - b * PP;
        const long orow = ((long)b * SS + 1 + p) * DD;
        const float sw = sumw[m];
        out[orow + col0] = c0[r] + sw * bv0;
        out[orow + col1] = c1[r] + sw * bv1;
    }
}

// ---------------------------------------------------------------------------
// Kernel 3: class-token row: out[b, 0, :] = 1.0  (attn==1, sv row 0 == ones)
// ---------------------------------------------------------------------------
__global__ void gsa_cls_ones(float* __restrict__ out)
{
    const int i = blockIdx.x * blockDim.x + threadIdx.x;  // 0 .. BB*DD-1
    const int b = i / DD;
    const int d = i - b * DD;
    out[(long)b * SS * DD + d] = 1.0f;
}

// ---------------------------------------------------------------------------
extern "C" void kernel_launch(void* const* d_in, const int* in_sizes, int n_in,
                              void* d_out, int out_size, void* d_ws, size_t ws_size,
                              hipStream_t stream)
{
    // setup_inputs order:
    // 0:x 1:img_ids 2:mask 3:Wq 4:bq 5:Wk 6:bk 7:Wv 8:bv 9:avgs 10:std_devs 11:noise
    const float* x       = (const float*)d_in[0];
    const int*   img_ids = (const int*)  d_in[1];
    const float* Wv      = (const float*)d_in[7];
    const float* bv      = (const float*)d_in[8];
    const float* avgs    = (const float*)d_in[9];
    const float* stds    = (const float*)d_in[10];
    const float* noise   = (const float*)d_in[11];
    float*       out     = (float*)d_out;

    // Workspace: [0, MM*4) sumw ; [MM*4 ...) xs (MM x DD f32)
    float* sumw = (float*)d_ws;
    float* xs   = (float*)((char*)d_ws + (size_t)MM * sizeof(float));

    gsa_sample_combine<<<MM, 256, 0, stream>>>(x, img_ids, avgs, stds, noise,
                                               xs, sumw);
    gsa_cls_ones<<<(BB * DD) / 256, 256, 0, stream>>>(out);
    dim3 grid(MM / BM, DD / BN);   // 196 x 12
    gsa_gemm_wmma<<<grid, 256, 0, stream>>>(xs, Wv, bv, sumw, out);
    (void)in_sizes; (void)n_in; (void)out_size; (void)ws_size;
}